// SpectralConv1dDHT_63342177682128
// MI455X (gfx1250) — compile-verified
//
#include <hip/hip_runtime.h>

// SpectralConv1dDHT on gfx1250 (MI455X, wave32, WMMA).
//
// out = (1/N) * Cas^T @ ( W ⊙-contract ( Cas @ x ) ), only first 64 modes.
// Stage1 & Stage3 are f32 WMMA GEMMs (v_wmma_f32_16x16x4_f32), stage2 is a
// tiny VALU contraction. Deterministic (fixed partitioning, fixed-order serial sum).

typedef __attribute__((ext_vector_type(2))) float v2f;
typedef __attribute__((ext_vector_type(8))) float v8f;

#define NN     16384
#define MM     64
#define CIN    64
#define COUT   64
#define BBATCH 16
#define ROWS   1024                 // B*Cin == B*Cout
#define S1_CHUNKS 32                // K split for stage-1 occupancy
#define S1_NPER   (NN / S1_CHUNKS)  // 512 n per chunk -> 128 k-steps

// workspace layout (floats):
#define WS_CAS   0                        // cas  [64][16384]   : 1,048,576
#define WS_CAST  (WS_CAS  + MM * NN)      // casT [16384][64]   : 1,048,576
#define WS_PART  (WS_CAST + NN * MM)      // part [32][1024][64]: 2,097,152
#define WS_XH    (WS_PART + S1_CHUNKS * ROWS * MM)  // Xh [1024][64]
#define WS_LOW   (WS_XH   + ROWS * MM)              // low[1024][64]
// total = 4,325,376 floats = 16.5 MiB of d_ws

static __device__ __forceinline__ v8f wmma4(v2f a, v2f b, v8f c) {
    // D = A(16x4,f32) * B(4x16,f32) + C(16x16,f32)
    return __builtin_amdgcn_wmma_f32_16x16x4_f32(
        /*neg_a=*/false, a, /*neg_b=*/false, b,
        /*c_mod=*/(short)0, c, /*reuse_a=*/false, /*reuse_b=*/false);
}

// ---------------------------------------------------------------- cas tables
__global__ void cas_build(float* __restrict__ cas, float* __restrict__ casT) {
    int idx = blockIdx.x * blockDim.x + threadIdx.x;   // 1,048,576 threads
    int n = idx & (NN - 1);
    int m = idx >> 14;
    int phase = (m * n) & (NN - 1);                    // exact mod-N in int
    float ang = (float)phase * (6.283185307179586f / (float)NN);
    float s, c;
    sincosf(ang, &s, &c);
    float v = c + s;                                   // cas()
    cas[m * NN + n]  = v;                              // coalesced
    casT[n * MM + m] = v;                              // one-time scatter
}

// ------------------------------------------- stage 1: Xh_part = x . cas^T
// one wave per (rowTile, kChunk); 4 mode-tiles (all 64 modes) per wave.
__global__ void stage1_fwd(const float* __restrict__ x,
                           const float* __restrict__ cas,
                           float* __restrict__ part) {
    int wave = (blockIdx.x * blockDim.x + threadIdx.x) >> 5;  // 2048 waves
    int lane = threadIdx.x & 31;
    int c  = wave & (S1_CHUNKS - 1);
    int rt = wave >> 5;                                // 0..63
    int half = lane >> 4;                              // K-half select
    int l15  = lane & 15;
    int koff = half * 2;                               // lanes>=16 hold K=2,3
    int rowBase = rt * 16;
    int n0 = c * S1_NPER;

    const float* ap  = x   + (size_t)(rowBase + l15) * NN + n0 + koff;  // A row
    const float* bp0 = cas + (size_t)( 0 + l15) * NN + n0 + koff;       // m tile 0
    const float* bp1 = cas + (size_t)(16 + l15) * NN + n0 + koff;       // m tile 1
    const float* bp2 = cas + (size_t)(32 + l15) * NN + n0 + koff;       // m tile 2
    const float* bp3 = cas + (size_t)(48 + l15) * NN + n0 + koff;       // m tile 3

    v8f acc0 = {}, acc1 = {}, acc2 = {}, acc3 = {};
#pragma unroll 4
    for (int k = 0; k < S1_NPER; k += 4) {
        v2f a  = *(const v2f*)(ap  + k);
        v2f b0 = *(const v2f*)(bp0 + k);
        v2f b1 = *(const v2f*)(bp1 + k);
        v2f b2 = *(const v2f*)(bp2 + k);
        v2f b3 = *(const v2f*)(bp3 + k);
        acc0 = wmma4(a, b0, acc0);      // 4 independent chains -> pipelined
        acc1 = wmma4(a, b1, acc1);
        acc2 = wmma4(a, b2, acc2);
        acc3 = wmma4(a, b3, acc3);
    }

    // D layout: lane<16 -> M=v, col=lane ; lane>=16 -> M=v+8, col=lane-16
    float* pb = part + ((size_t)c * ROWS + rowBase + half * 8) * MM + l15;
#pragma unroll
    for (int v = 0; v < 8; ++v) {
        pb[(size_t)v * MM +  0] = acc0[v];
        pb[(size_t)v * MM + 16] = acc1[v];
        pb[(size_t)v * MM + 32] = acc2[v];
        pb[(size_t)v * MM + 48] = acc3[v];
    }
}

// ---------------------------------------------- stage 1b: deterministic sum
__global__ void stage1_reduce(const float* __restrict__ part,
                              float* __restrict__ Xh) {
    int idx = blockIdx.x * blockDim.x + threadIdx.x;   // 65,536 threads
    float acc = 0.0f;
#pragma unroll
    for (int c = 0; c < S1_CHUNKS; ++c)
        acc += part[(size_t)c * (ROWS * MM) + idx];
    Xh[idx] = acc;
}

// -------------------------- stage 2: low[b,o,m] = sum_i Xh[b,i,m]*W[o,i,m]
__global__ void stage2_mix(const float* __restrict__ Xh,
                           const float* __restrict__ w,
                           float* __restrict__ low) {
    int idx = blockIdx.x * blockDim.x + threadIdx.x;   // 65,536 threads
    int m = idx & 63;
    int o = (idx >> 6) & 63;
    int b = idx >> 12;
    const float* xh = Xh + (size_t)b * CIN * MM + m;   // stride MM over i
    const float* wp = w  + (size_t)o * CIN * MM + m;
    float acc = 0.0f;
#pragma unroll 8
    for (int i = 0; i < CIN; ++i)
        acc = fmaf(xh[i * MM], wp[i * MM], acc);
    low[((size_t)b * COUT + o) * MM + m] = acc;        // row-major (b*64+o, m)
}

// ------------------------------- stage 3: out = (1/N) * low . cas (K = 64)
// one wave per (rowTile, 64-wide n group); A preloaded, 2 interleaved accs.
__global__ void stage3_inv(const float* __restrict__ low,
                           const float* __restrict__ casT,
                           float* __restrict__ out) {
    int wave = (blockIdx.x * blockDim.x + threadIdx.x) >> 5;  // 16384 waves
    int lane = threadIdx.x & 31;
    int ng = wave & 255;                               // n group: 64 columns
    int rt = wave >> 8;                                // 0..63
    int half = lane >> 4;
    int l15  = lane & 15;
    int koff = half * 2;
    int rowBase = rt * 16;

    // preload all 16 A fragments (K=64) — contiguous in m
    v2f afrag[16];
    const float* lrow = low + (size_t)(rowBase + l15) * MM + koff;
#pragma unroll
    for (int s = 0; s < 16; ++s) afrag[s] = *(const v2f*)(lrow + 4 * s);

    const float scale = 1.0f / (float)NN;
    int nBase0 = ng * 64;
    float* ob = out + (size_t)(rowBase + half * 8) * NN + l15;

    for (int p = 0; p < 2; ++p) {
        int nA = nBase0 + p * 32;
        int nB = nA + 16;
        const float* ctA = casT + (size_t)(nA + l15) * MM + koff;
        const float* ctB = casT + (size_t)(nB + l15) * MM + koff;
        v8f acc0 = {}, acc1 = {};
#pragma unroll
        for (int s = 0; s < 16; ++s) {
            v2f b0 = *(const v2f*)(ctA + 4 * s);
            v2f b1 = *(const v2f*)(ctB + 4 * s);
            acc0 = wmma4(afrag[s], b0, acc0);
            acc1 = wmma4(afrag[s], b1, acc1);
        }
#pragma unroll
        for (int v = 0; v < 8; ++v) {
            ob[(size_t)v * NN + nA] = acc0[v] * scale;
            ob[(size_t)v * NN + nB] = acc1[v] * scale;
        }
    }
}

// ---------------------------------------------------------------- launcher
extern "C" void kernel_launch(void* const* d_in, const int* in_sizes, int n_in,
                              void* d_out, int out_size, void* d_ws, size_t ws_size,
                              hipStream_t stream) {
    const float* x = (const float*)d_in[0];        // [16, 64, 16384]
    const float* w = (const float*)d_in[1];        // [64, 64, 64]
    // d_in[2] = modes1 (== 64, hardcoded)
    float* out = (float*)d_out;                    // [16, 64, 16384]

    float* ws   = (float*)d_ws;                    // needs 16.5 MiB
    float* cas  = ws + WS_CAS;
    float* casT = ws + WS_CAST;
    float* part = ws + WS_PART;
    float* Xh   = ws + WS_XH;
    float* low  = ws + WS_LOW;

    cas_build    <<<(MM * NN) / 256, 256, 0, stream>>>(cas, casT);
    stage1_fwd   <<<(64 * S1_CHUNKS) / 4, 128, 0, stream>>>(x, cas, part);   // 2048 waves
    stage1_reduce<<<(ROWS * MM) / 256, 256, 0, stream>>>(part, Xh);
    stage2_mix   <<<(BBATCH * COUT * MM) / 256, 256, 0, stream>>>(Xh, w, low);
    stage3_inv   <<<(64 * 256) / 4, 128, 0, stream>>>(low, casT, out);       // 16384 waves
}